// GatedFusionTokens_88734024335568
// MI455X (gfx1250) — compile-verified
//
#include <hip/hip_runtime.h>

typedef float v2f __attribute__((ext_vector_type(2)));
typedef float v8f __attribute__((ext_vector_type(8)));

#define NLOG2E (-1.44269504088896340736f)

// Fixed problem shape from the reference (T derived host-side).
constexpr int H  = 256;                 // hidden dim
constexpr int E  = 128;                 // events
constexpr int EH = E * H;               // gates row stride (elements)

// ---------------------------------------------------------------------------
// Kernel 1: te'[e,h] = -log2e * (event_embeds @ W[H:] + b)[e,h]
// One wave per 16x16 tile of the [E,H] result, f32 WMMA over K=H.
// ---------------------------------------------------------------------------
__global__ void __launch_bounds__(256)
te_prep_kernel(const float* __restrict__ ev, const float* __restrict__ W,
               const float* __restrict__ b, float* __restrict__ tep) {
    const int wave = (blockIdx.x * blockDim.x + threadIdx.x) >> 5;
    const int lane = threadIdx.x & 31;
    constexpr int ntile = H >> 4;                // tiles along H
    const int e0 = (wave / ntile) << 4;
    const int n0 = (wave % ntile) << 4;
    const int lm = lane & 15;                    // A-row m / B-col n / D-col n
    const int hi = lane >> 4;                    // K-pair select / D row half

    // Lane-invariant-in-loop base pointers: all k-loop loads use imm offsets.
    const float* __restrict__ aptr = ev + (size_t)(e0 + lm) * H + 2 * hi;
    const float* __restrict__ bptr = W + (size_t)H * H + n0 + lm + (size_t)(2 * hi) * H;

    v8f c = {};
    #pragma unroll 8
    for (int k = 0; k < H; k += 4) {
        v2f a = *(const v2f*)(aptr + k);         // A[m, k+2hi], A[m, k+2hi+1]
        v2f bb;
        bb.x = bptr[(size_t)k * H];              // B[k+2hi,   n]
        bb.y = bptr[(size_t)k * H + H];          // B[k+2hi+1, n]
        c = __builtin_amdgcn_wmma_f32_16x16x4_f32(false, a, false, bb,
                                                  (short)0, c, false, false);
    }

    const float bias = b[n0 + lm];
    float* __restrict__ out = tep + (size_t)(e0 + 8 * hi) * H + n0 + lm;
    #pragma unroll
    for (int r = 0; r < 8; ++r)
        out[r * H] = NLOG2E * (c[r] + bias);     // pre-negated & log2-scaled
}

// ---------------------------------------------------------------------------
// Kernel 2: per 16x16 (t,h) tile:
//   tn' = -log2e * (note @ W[:H])   via f32 WMMA
//   for e: g = 1/(1+exp2(tn' + te')); store gates; acc += g*event
//   enriched = note + acc
// ---------------------------------------------------------------------------
__global__ void __launch_bounds__(128)
gated_fusion_kernel(const float* __restrict__ note, const float* __restrict__ ev,
                    const float* __restrict__ W, const float* __restrict__ tep,
                    float* __restrict__ enriched, float* __restrict__ gates) {
    const int lane = threadIdx.x & 31;
    const int t0   = blockIdx.x << 4;
    const int n0   = (blockIdx.y << 6) + ((threadIdx.x >> 5) << 4);
    const int lm   = lane & 15;
    const int hi   = lane >> 4;

    // ---- tn tile via WMMA (K = H, step 4), imm-offset loads only ----
    const float* __restrict__ aptr = note + (size_t)(t0 + lm) * H + 2 * hi;
    const float* __restrict__ bptr = W + n0 + lm + (size_t)(2 * hi) * H;

    v8f c = {};
    #pragma unroll 8
    for (int k = 0; k < H; k += 4) {
        v2f a = *(const v2f*)(aptr + k);
        v2f bb;
        bb.x = bptr[(size_t)k * H];
        bb.y = bptr[(size_t)k * H + H];
        c = __builtin_amdgcn_wmma_f32_16x16x4_f32(false, a, false, bb,
                                                  (short)0, c, false, false);
    }

    float xln[8], acc[8];
    #pragma unroll
    for (int r = 0; r < 8; ++r) { xln[r] = NLOG2E * c[r]; acc[r] = 0.0f; }

    // ---- streaming gate pass: the bandwidth-dominant part ----
    const float* __restrict__ tp = tep + n0 + lm;            // te'[e, col]
    const float* __restrict__ ep = ev  + n0 + lm;            // event[e, col]
    float* __restrict__ gbase = gates + (size_t)(t0 + 8 * hi) * EH + n0 + lm;

    #pragma unroll 4
    for (int e = 0; e < E; ++e) {
        const float te_s = tp[e * H];
        const float ev_s = ep[e * H];
        float* __restrict__ g = gbase + (size_t)e * H;
        #pragma unroll
        for (int r = 0; r < 8; ++r) {
            // sigmoid(logit) = 1 / (1 + exp2(-logit*log2e)); inputs pre-scaled
            const float z  = __builtin_amdgcn_exp2f(xln[r] + te_s);
            const float gv = __builtin_amdgcn_rcpf(1.0f + z);
            g[r * EH] = gv;                                   // imm-offset stores
            acc[r] = __builtin_fmaf(gv, ev_s, acc[r]);
        }
    }

    // ---- enriched = note + enrichment ----
    const float* __restrict__ nrow = note + (size_t)(t0 + 8 * hi) * H + n0 + lm;
    float* __restrict__ orow = enriched + (size_t)(t0 + 8 * hi) * H + n0 + lm;
    #pragma unroll
    for (int r = 0; r < 8; ++r)
        orow[r * H] = nrow[r * H] + acc[r];
}

// ---------------------------------------------------------------------------
extern "C" void kernel_launch(void* const* d_in, const int* in_sizes, int n_in,
                              void* d_out, int out_size, void* d_ws, size_t ws_size,
                              hipStream_t stream) {
    const float* note = (const float*)d_in[0];
    const float* ev   = (const float*)d_in[1];
    const float* W    = (const float*)d_in[2];
    const float* b    = (const float*)d_in[3];

    const int T = in_sizes[0] / H;        // 4096

    float* enriched = (float*)d_out;
    float* gatesOut = (float*)d_out + (size_t)T * H;
    float* tep      = (float*)d_ws;       // E*H floats = 128 KB scratch

    // Kernel 1: (E/16)*(H/16) = 128 waves -> 16 blocks x 256 threads (8 waves)
    constexpr int wavesA  = (E / 16) * (H / 16);
    constexpr int blocksA = (wavesA * 32 + 255) / 256;
    te_prep_kernel<<<blocksA, 256, 0, stream>>>(ev, W, b, tep);

    // Kernel 2: grid (T/16, H/64), 128-thread (4-wave) blocks
    dim3 grd(T / 16, H / 64);
    gated_fusion_kernel<<<grd, 128, 0, stream>>>(note, ev, W, tep,
                                                 enriched, gatesOut);
}